// AttentionBlock_56075093016781
// MI455X (gfx1250) — compile-verified
//
#include <hip/hip_runtime.h>
#include <math.h>

// CDNA5 fp32 WMMA fragments (wave32):
//  A 16x4 : per-lane float2 = A[m][k0], A[m][k0+1], m = lane&15, k0 = 2*(lane>>4)
//  B 4x16 : per-lane float2 = B[k0][n], B[k0+1][n],  n = lane&15, k0 = 2*(lane>>4)
//  C/D 16x16 : 8 VGPRs, VGPR i -> (M = i + 8*(lane>>4), N = lane&15)
typedef __attribute__((ext_vector_type(2))) float v2f;
typedef __attribute__((ext_vector_type(8))) float v8f;

#define BN_EPS 1e-5f

__device__ __forceinline__ v8f wmma_f32_k4(v2f a, v2f b, v8f c) {
  // D = A(16x4) * B(4x16) + C, full fp32
  return __builtin_amdgcn_wmma_f32_16x16x4_f32(
      /*neg_a=*/false, a, /*neg_b=*/false, b,
      /*c_mod=*/(short)0, c, /*reuse_a=*/false, /*reuse_b=*/false);
}

// ---------------------------------------------------------------------------
// Kernel 1: BatchNorm1d (training stats over batch*length per channel).
// One block per channel; 2048 elements reduced per block.
// ---------------------------------------------------------------------------
__global__ void bn_normalize(const float* __restrict__ x,
                             const float* __restrict__ gamma,
                             const float* __restrict__ beta,
                             float* __restrict__ xn,
                             int B, int C, int L) {
  const int ch = blockIdx.x;
  const int tid = threadIdx.x;
  const int n = B * L;
  __shared__ float s_sum[8];
  __shared__ float s_sq[8];

  float s = 0.f, sq = 0.f;
  for (int i = tid; i < n; i += blockDim.x) {
    const int b = i >> 10;        // L = 1024
    const int l = i & (L - 1);
    const float v = x[((size_t)b * C + ch) * L + l];
    s += v;
    sq += v * v;
  }
  for (int off = 16; off > 0; off >>= 1) {
    s  += __shfl_xor(s, off, 32);
    sq += __shfl_xor(sq, off, 32);
  }
  const int wave = tid >> 5;
  const int lane = tid & 31;
  if (lane == 0) { s_sum[wave] = s; s_sq[wave] = sq; }
  __syncthreads();
  if (tid == 0) {
    float ts = 0.f, tq = 0.f;
    const int nw = blockDim.x >> 5;
    for (int w = 0; w < nw; ++w) { ts += s_sum[w]; tq += s_sq[w]; }
    s_sum[0] = ts; s_sq[0] = tq;
  }
  __syncthreads();
  const float inv_n = 1.0f / (float)n;
  const float mean = s_sum[0] * inv_n;
  const float var  = s_sq[0] * inv_n - mean * mean;  // biased variance
  const float g = gamma[ch] * rsqrtf(var + BN_EPS);
  const float bt = beta[ch];
  for (int i = tid; i < n; i += blockDim.x) {
    const int b = i >> 10;
    const int l = i & (L - 1);
    const size_t idx = ((size_t)b * C + ch) * L + l;
    xn[idx] = (x[idx] - mean) * g + bt;
  }
}

// ---------------------------------------------------------------------------
// Kernel 2: fp32 WMMA GEMM  C[z] = A (MxK) * B[z] (KxN) + bias (+ resid[z])
// One wave per 16x16 output tile; K-loop of v_wmma_f32_16x16x4_f32.
// ---------------------------------------------------------------------------
__global__ void gemm_wmma_f32(const float* __restrict__ A,     // M x K
                              const float* __restrict__ Bm,    // [z] K x N
                              const float* __restrict__ bias,  // M
                              const float* __restrict__ resid, // [z] M x N or null
                              float* __restrict__ C,           // [z] M x N
                              int M, int N, int K) {
  const int lane = threadIdx.x;
  const int n15  = lane & 15;
  const int hi   = lane >> 4;
  const int nt = blockIdx.x * 16;
  const int mt = blockIdx.y * 16;
  const int bz = blockIdx.z;

  const float* Bp = Bm + (size_t)bz * K * N;
  float*       Cp = C  + (size_t)bz * M * N;
  const float* Rp = resid ? (resid + (size_t)bz * M * N) : nullptr;

  const float* arow = A  + (size_t)(mt + n15) * K + 2 * hi;  // -> A[m][2hi + k]
  const float* bcol = Bp + (size_t)(2 * hi) * N + nt + n15;  // -> B[2hi + k][n]

  v8f acc = {};
#pragma unroll 8
  for (int k = 0; k < K; k += 4) {
    v2f a = *(const v2f*)(arow + k);
    v2f b;
    b.x = bcol[(size_t)k * N];
    b.y = bcol[(size_t)k * N + N];
    acc = wmma_f32_k4(a, b, acc);
  }

#pragma unroll
  for (int i = 0; i < 8; ++i) {
    const int row = mt + i + 8 * hi;
    const size_t idx = (size_t)row * N + nt + n15;
    float v = acc[i] + bias[row];
    if (Rp) v += Rp[idx];
    Cp[idx] = v;
  }
}

// ---------------------------------------------------------------------------
// Kernel 3: fused attention for the quirky layout.
//   q[b,h,t,j] = qkv[b, j*hd + h,      t]   (j in 0..nh-1, h in 0..hd-1)
//   k[b,h,f,j] = qkv[b, (nh+j)*hd + h, f]
//   v[b,h,f,j] = qkv[b, (2nh+j)*hd+h,  f]
//   S = softmax_f( scale * Q K^T ),  O = S V,  h_out[b, j*hd+h, t] = O[t,j]
// One wave per (b, h, 16-row tile). 16x1024 score strip lives in LDS
// (row pad +4 dwords -> conflict-free lanes-by-row reads in phase 3).
// ---------------------------------------------------------------------------
#define SROW 1028   // 1024 + 4 floats padding

__global__ void attn_wmma(const float* __restrict__ qkv,  // [b][3*c][L]
                          float* __restrict__ hbuf,       // [b][c][L]
                          int L, int nh, int hd) {
  __shared__ float sS[16 * SROW];   // ~65.8 KB of the 320 KB/WGP LDS

  const int lane = threadIdx.x;
  const int n15  = lane & 15;
  const int hi   = lane >> 4;
  const int t0 = blockIdx.x * 16;
  const int h  = blockIdx.y;
  const int bz = blockIdx.z;

  const size_t chanstride = (size_t)hd * L;                        // 65536
  const float* base = qkv + (size_t)bz * 3 * nh * chanstride;      // batch base
  const float scale = rsqrtf((float)(3 * nh));

  // ---- Phase 1: S[t0+m][f] = scale * sum_j Q[t,j] K[f,j] via WMMA ----
  const float* qp = base + (size_t)h * L + t0 + n15;               // + j*chanstride
  const float* kp = base + (size_t)nh * chanstride + (size_t)h * L + n15;
  for (int f0 = 0; f0 < L; f0 += 16) {
    v8f acc = {};
#pragma unroll
    for (int k0 = 0; k0 < 8; k0 += 4) {        // K = nh = 8 -> 2 WMMAs
      const int kk = k0 + 2 * hi;
      v2f a, b;
      a.x = qp[(size_t)kk * chanstride] * scale;
      a.y = qp[(size_t)(kk + 1) * chanstride] * scale;
      b.x = kp[(size_t)kk * chanstride + f0];
      b.y = kp[(size_t)(kk + 1) * chanstride + f0];
      acc = wmma_f32_k4(a, b, acc);
    }
#pragma unroll
    for (int i = 0; i < 8; ++i)
      sS[(i + 8 * hi) * SROW + f0 + n15] = acc[i];
  }
  __syncthreads();

  // ---- Phase 2: row softmax over L, wave32-cooperative, in LDS ----
  for (int m = 0; m < 16; ++m) {
    float mx = -3.4e38f;
    for (int f = lane; f < L; f += 32) mx = fmaxf(mx, sS[m * SROW + f]);
    for (int off = 16; off > 0; off >>= 1) mx = fmaxf(mx, __shfl_xor(mx, off, 32));
    float sum = 0.f;
    for (int f = lane; f < L; f += 32) {
      const float e = __expf(sS[m * SROW + f] - mx);
      sS[m * SROW + f] = e;
      sum += e;
    }
    for (int off = 16; off > 0; off >>= 1) sum += __shfl_xor(sum, off, 32);
    const float inv = 1.0f / sum;
    for (int f = lane; f < L; f += 32) sS[m * SROW + f] *= inv;
  }
  __syncthreads();

  // ---- Phase 3: O(16 x nh) = P(16 x L) * V(L x nh), N padded to 16 ----
  const float* vcol = base + (size_t)(2 * nh) * chanstride
                           + (size_t)n15 * chanstride + (size_t)h * L;  // V[:, n]
  v8f acc = {};
  for (int k0 = 0; k0 < L; k0 += 4) {
    const int kk = k0 + 2 * hi;
    v2f a = *(const v2f*)&sS[n15 * SROW + kk];   // P rows straight from LDS
    v2f b;
    if (n15 < nh) { b.x = vcol[kk]; b.y = vcol[kk + 1]; }
    else          { b.x = 0.f;      b.y = 0.f; }
    acc = wmma_f32_k4(a, b, acc);
  }

  // h_out[b, n*hd + h, t0+m] = O[m][n]  (only n < nh valid)
  float* hb = hbuf + (size_t)bz * (size_t)nh * chanstride;
  if (n15 < nh) {
#pragma unroll
    for (int i = 0; i < 8; ++i) {
      const int m = i + 8 * hi;
      hb[((size_t)n15 * hd + h) * L + t0 + m] = acc[i];
    }
  }
}

// ---------------------------------------------------------------------------
extern "C" void kernel_launch(void* const* d_in, const int* in_sizes, int n_in,
                              void* d_out, int out_size, void* d_ws, size_t ws_size,
                              hipStream_t stream) {
  (void)in_sizes; (void)n_in; (void)out_size; (void)ws_size;

  const float* x      = (const float*)d_in[0];
  const float* gamma  = (const float*)d_in[1];
  const float* beta   = (const float*)d_in[2];
  const float* w_qkv  = (const float*)d_in[3];
  const float* b_qkv  = (const float*)d_in[4];
  const float* w_proj = (const float*)d_in[5];
  const float* b_proj = (const float*)d_in[6];
  // d_in[7] = num_heads (fixed 8 for this problem shape)

  const int B = 2, C = 512, L = 1024, NH = 8, HD = 64;

  // Workspace layout (fp32): xn (4 MiB) | qkv (12 MiB) | h (4 MiB)
  float* xn   = (float*)d_ws;
  float* qkv  = xn  + (size_t)B * C * L;
  float* hbuf = qkv + (size_t)B * 3 * C * L;
  float* out  = (float*)d_out;

  // 1) BatchNorm (training-mode stats, biased var)
  bn_normalize<<<dim3(C), dim3(256), 0, stream>>>(x, gamma, beta, xn, B, C, L);

  // 2) QKV projection: qkv[b] = w_qkv (1536x512) * xn[b] (512x1024) + b_qkv
  gemm_wmma_f32<<<dim3(L / 16, (3 * C) / 16, B), dim3(32), 0, stream>>>(
      w_qkv, xn, b_qkv, nullptr, qkv, 3 * C, L, C);

  // 3) Fused attention (scores + softmax + PV) per (b, h, t-tile)
  attn_wmma<<<dim3(L / 16, HD, B), dim3(32), 0, stream>>>(qkv, hbuf, L, NH, HD);

  // 4) Output projection + residual: out[b] = x[b] + w_proj * h[b] + b_proj
  gemm_wmma_f32<<<dim3(L / 16, C / 16, B), dim3(32), 0, stream>>>(
      w_proj, hbuf, b_proj, x, out, C, L, C);
}